// GraphSageModel_11836929868177
// MI455X (gfx1250) — compile-verified
//
#include <hip/hip_runtime.h>
#include <hip/hip_bf16.h>

// ---------------------------------------------------------------------------
// GraphSAGE (gcn aggregator) forward for MI455X / gfx1250.
//
// Phases:
//   1) deg[i]      = sum over edges with dst==i of 1           (fp32 atomics)
//   2) agg[i,:]   += h[src,:] for each edge (dst scatter)      (fp32 atomics, L2-resident)
//   3) agg        = (agg + h) / (deg + 1)                      (elementwise, in place)
//   4) h' = act(agg @ W + b)  via V_WMMA_F32_16X16X4_F32       (exact fp32 matrix path)
// repeated twice, then two dense layers with LeakyReLU(0.01).
// ---------------------------------------------------------------------------

#define N_NODES   50000
#define N_EDGES   800000
#define FEATS     128
#define HIDDEN    128
#define DNN1      256
#define DNN2      128

typedef float v2f __attribute__((ext_vector_type(2)));
typedef float v8f __attribute__((ext_vector_type(8)));

__device__ __forceinline__ void atomic_add_f32(float* p, float v) {
    // Lowers to GLOBAL_ATOMIC_ADD_F32 (non-returning, agent scope).
    __hip_atomic_fetch_add(p, v, __ATOMIC_RELAXED, __HIP_MEMORY_SCOPE_AGENT);
}

// --------------------------- degree ----------------------------------------
__global__ void __launch_bounds__(256)
deg_kernel(const int* __restrict__ dst, float* __restrict__ deg, int nEdges) {
    int e = blockIdx.x * 256 + threadIdx.x;
    if (e < nEdges) atomic_add_f32(&deg[dst[e]], 1.0f);
}

// --------------------------- edge scatter ----------------------------------
// One wave (32 lanes) per edge; each lane moves 4 consecutive floats.
// 800K * 128 floats of gather + atomic-add; whole working set (25.6MB) is
// L2-resident on the 192MB L2, so hardware fp32 atomics are the fast path.
__global__ void __launch_bounds__(256)
edge_scatter(const float* __restrict__ H, const int* __restrict__ src,
             const int* __restrict__ dst, float* __restrict__ agg, int nEdges) {
    long gid  = (long)blockIdx.x * 256 + threadIdx.x;
    long e    = gid >> 5;
    if (e >= nEdges) return;
    int  c    = ((int)gid & 31) * 4;
    long s    = src[e];
    long d    = dst[e];
    const float4 v = *(const float4*)(H + s * (long)FEATS + c);
    float* out = agg + d * (long)FEATS + c;
    atomic_add_f32(out + 0, v.x);
    atomic_add_f32(out + 1, v.y);
    atomic_add_f32(out + 2, v.z);
    atomic_add_f32(out + 3, v.w);
}

// --------------------------- normalize (in place) --------------------------
// agg[i,:] = (agg[i,:] + h[i,:]) / (deg[i] + 1)
__global__ void __launch_bounds__(256)
normalize_kernel(float* __restrict__ agg, const float* __restrict__ h,
                 const float* __restrict__ deg, int nNodes) {
    long gid = (long)blockIdx.x * 256 + threadIdx.x;         // one float4 per thread
    long total = (long)nNodes * (FEATS / 4);
    if (gid >= total) return;
    long node = gid / (FEATS / 4);
    float inv = 1.0f / (deg[node] + 1.0f);
    float4 a  = ((const float4*)agg)[gid];
    float4 hv = ((const float4*)h)[gid];
    a.x = (a.x + hv.x) * inv;
    a.y = (a.y + hv.y) * inv;
    a.z = (a.z + hv.z) * inv;
    a.w = (a.w + hv.w) * inv;
    ((float4*)agg)[gid] = a;
}

// --------------------------- WMMA fp32 GEMM + bias + activation ------------
// Y[M,N] = act(X[M,K] @ W[K,N] + b[N]),  act: 0 = ReLU, 1 = LeakyReLU(0.01)
//
// Block: 128 threads = 4 waves. Block covers 16 rows x 64 cols; each wave
// owns one 16x16 tile, accumulating K/4 V_WMMA_F32_16X16X4_F32 ops.
// A-tile (16 x K) staged in LDS with +4-float row padding to spread the
// per-M fragment reads across banks. B fragments read straight from global:
// W is tiny (<=128KB) and hammered by 3125 blocks -> pure L2 hits.
//
// Fragment layouts per CDNA5 ISA Ch7.12.2:
//   A 16x4 fp32 : lane = M (0-15 / 16-31 same M), VGPR0/1 = {k, k+1},
//                 lanes 16-31 hold k+2 / k+3.
//   B 4x16 fp32 : lane = N, same K split across the register pair.
//   C/D 16x16   : VGPR v = row v (lanes 0-15) and row v+8 (lanes 16-31).
__global__ void __launch_bounds__(128)
gemm_bias_act(const float* __restrict__ X, const float* __restrict__ W,
              const float* __restrict__ bias, float* __restrict__ Y,
              int M, int K, int N, int act) {
    extern __shared__ float alds[];                 // 16 * (K + 4) floats
    const int lda   = K + 4;
    const int tid   = threadIdx.x;
    const int lane  = tid & 31;
    const int wave  = tid >> 5;
    const long rowBase = (long)blockIdx.x * 16;
    const int col0  = (blockIdx.y * 4 + wave) * 16;

    // Stage the 16 x K A-tile into LDS (coalesced float4 global loads).
    for (int idx = tid; idx < (16 * K) / 4; idx += 128) {
        int r = (idx * 4) / K;
        int c = (idx * 4) % K;
        const float4 v = *(const float4*)(X + (rowBase + r) * K + c);
        float* p = alds + r * lda + c;
        p[0] = v.x; p[1] = v.y; p[2] = v.z; p[3] = v.w;
    }
    __syncthreads();

    const int half = lane >> 4;      // 0: K offset 0, 1: K offset 2
    const int mrow = lane & 15;      // M for A fragment, N for B fragment
    v8f acc = {};
    for (int k0 = 0; k0 < K; k0 += 4) {
        const int kk = k0 + 2 * half;
        v2f a;
        a.x = alds[mrow * lda + kk];
        a.y = alds[mrow * lda + kk + 1];
        v2f b;
        const float* wp = W + (long)kk * N + col0 + mrow;
        b.x = wp[0];
        b.y = wp[N];
        acc = __builtin_amdgcn_wmma_f32_16x16x4_f32(
            /*neg_a=*/false, a, /*neg_b=*/false, b,
            /*c_mod=*/(short)0, acc, /*reuse_a=*/false, /*reuse_b=*/false);
    }

    const int col = col0 + mrow;
    const float bv = bias[col];
    #pragma unroll
    for (int v = 0; v < 8; ++v) {
        long row = rowBase + v + 8 * half;
        if (row < M) {
            float y = acc[v] + bv;
            y = (act == 0) ? fmaxf(y, 0.0f) : (y > 0.0f ? y : 0.01f * y);
            Y[row * N + col] = y;
        }
    }
}

// --------------------------- host side -------------------------------------
extern "C" void kernel_launch(void* const* d_in, const int* in_sizes, int n_in,
                              void* d_out, int out_size, void* d_ws, size_t ws_size,
                              hipStream_t stream) {
    const float* features = (const float*)d_in[0];
    const int*   src      = (const int*)  d_in[1];
    const int*   dst      = (const int*)  d_in[2];
    const float* w1       = (const float*)d_in[3];
    const float* b1       = (const float*)d_in[4];
    const float* w2       = (const float*)d_in[5];
    const float* b2       = (const float*)d_in[6];
    const float* dw1      = (const float*)d_in[7];
    const float* db1      = (const float*)d_in[8];
    const float* dw2      = (const float*)d_in[9];
    const float* db2      = (const float*)d_in[10];
    float*       out      = (float*)d_out;

    // Workspace layout (floats): deg | agg | hbuf | t256   (~102.6 MB total)
    float* ws   = (float*)d_ws;
    float* deg  = ws;                                        // 50048
    float* agg  = ws + 50048;                                // 50000*128
    float* hbuf = agg + (size_t)N_NODES * FEATS;             // 50000*128
    float* t256 = hbuf + (size_t)N_NODES * FEATS;            // 50000*256

    const int edgeBlocks    = (N_EDGES + 255) / 256;
    const int scatterBlocks = (int)(((long)N_EDGES * 32 + 255) / 256);
    const int normBlocks    = (int)(((long)N_NODES * (FEATS / 4) + 255) / 256);

    // ---- degree ----
    hipMemsetAsync(deg, 0, 50048 * sizeof(float), stream);
    deg_kernel<<<edgeBlocks, 256, 0, stream>>>(dst, deg, N_EDGES);

    // ---- layer 1: aggregate features -> agg, normalize, GEMM(w1)+ReLU -> hbuf
    hipMemsetAsync(agg, 0, (size_t)N_NODES * FEATS * sizeof(float), stream);
    edge_scatter<<<scatterBlocks, 256, 0, stream>>>(features, src, dst, agg, N_EDGES);
    normalize_kernel<<<normBlocks, 256, 0, stream>>>(agg, features, deg, N_NODES);
    {
        dim3 grid(N_NODES / 16, HIDDEN / 64);
        size_t shmem = 16 * (FEATS + 4) * sizeof(float);
        gemm_bias_act<<<grid, 128, shmem, stream>>>(agg, w1, b1, hbuf,
                                                    N_NODES, FEATS, HIDDEN, /*relu*/0);
    }

    // ---- layer 2: aggregate hbuf -> agg, normalize, GEMM(w2)+ReLU -> agg? no:
    // write h2 into hbuf's slot after agg is consumed; reuse: GEMM reads agg,
    // writes over hbuf (its input h was already folded into agg by normalize).
    hipMemsetAsync(agg, 0, (size_t)N_NODES * FEATS * sizeof(float), stream);
    edge_scatter<<<scatterBlocks, 256, 0, stream>>>(hbuf, src, dst, agg, N_EDGES);
    normalize_kernel<<<normBlocks, 256, 0, stream>>>(agg, hbuf, deg, N_NODES);
    {
        dim3 grid(N_NODES / 16, HIDDEN / 64);
        size_t shmem = 16 * (HIDDEN + 4) * sizeof(float);
        gemm_bias_act<<<grid, 128, shmem, stream>>>(agg, w2, b2, hbuf,
                                                    N_NODES, HIDDEN, HIDDEN, /*relu*/0);
    }

    // ---- DNN head: Linear(128->256)+LeakyReLU, Linear(256->128)+LeakyReLU
    {
        dim3 grid(N_NODES / 16, DNN1 / 64);
        size_t shmem = 16 * (HIDDEN + 4) * sizeof(float);
        gemm_bias_act<<<grid, 128, shmem, stream>>>(hbuf, dw1, db1, t256,
                                                    N_NODES, HIDDEN, DNN1, /*leaky*/1);
    }
    {
        dim3 grid(N_NODES / 16, DNN2 / 64);
        size_t shmem = 16 * (DNN1 + 4) * sizeof(float);
        gemm_bias_act<<<grid, 128, shmem, stream>>>(t256, dw2, db2, out,
                                                    N_NODES, DNN1, DNN2, /*leaky*/1);
    }
}